// GraphVAE_5626407158311
// MI455X (gfx1250) — compile-verified
//
#include <hip/hip_runtime.h>
#include <math.h>

// ---------------------------------------------------------------------------
// GraphVAE on MI455X (gfx1250).  pseudo==0 => spline conv degenerates to
//   out = segment_mean(x[src], dst) @ W[0] + x @ root + b
// so we do edge scatter-adds (L2-resident) + node-tile WMMA f32 GEMMs.
// ---------------------------------------------------------------------------

#define N_NODES 50000
#define N_EDGES 800000
#define TILES   3125      // N_NODES / 16
#define WPB     5         // waves per block for GEMM kernels (3125 = 625*5)
#define GEMM_BLOCKS 625
#define GEMM_THREADS (WPB * 32)

typedef float v2f __attribute__((ext_vector_type(2)));
typedef float v8f __attribute__((ext_vector_type(8)));

static __device__ __forceinline__ v8f wmma4(v2f a, v2f b, v8f c) {
  // D = A(16x4,f32) * B(4x16,f32) + C(16x16,f32)
  return __builtin_amdgcn_wmma_f32_16x16x4_f32(
      /*neg_a=*/false, a, /*neg_b=*/false, b,
      /*c_mod=*/(short)0, c, /*reuse_a=*/false, /*reuse_b=*/false);
}

// --------------------------- utility kernels -------------------------------

__global__ void zero_kernel(float4* __restrict__ p, int n4) {
  int i = blockIdx.x * blockDim.x + threadIdx.x;
  if (i < n4) p[i] = make_float4(0.f, 0.f, 0.f, 0.f);
}

__global__ void deg_kernel(const int* __restrict__ dst, float* __restrict__ deg) {
  int e = blockIdx.x * blockDim.x + threadIdx.x;
  if (e < N_EDGES) atomicAdd(&deg[dst[e]], 1.0f);
}

// One thread per (edge, 4-feature chunk): 800k * 16 threads, float4 gather +
// 4 f32 atomic adds.  x / h are 12.8MB -> everything lives in the 192MB L2.
__global__ void scatter_kernel(const float* __restrict__ feat,
                               const int* __restrict__ src,
                               const int* __restrict__ dst,
                               float* __restrict__ agg) {
  int gid = blockIdx.x * blockDim.x + threadIdx.x;
  if (gid >= N_EDGES * 16) return;
  int e = gid >> 4;
  int c = (gid & 15) << 2;
  int s = src[e];
  int d = dst[e];
  float4 v = *reinterpret_cast<const float4*>(feat + (size_t)s * 64 + c);
  float* p = agg + (size_t)d * 64 + c;
  atomicAdd(p + 0, v.x);
  atomicAdd(p + 1, v.y);
  atomicAdd(p + 2, v.z);
  atomicAdd(p + 3, v.w);
}

// --------------------------- conv layer 1 ----------------------------------
// h = relu( (agg1/deg) @ W(64x64) + x @ R(64x64) + b )
__global__ void layer1_kernel(const float* __restrict__ x,
                              const float* __restrict__ agg,
                              const float* __restrict__ deg,
                              const float* __restrict__ W,
                              const float* __restrict__ R,
                              const float* __restrict__ b,
                              float* __restrict__ h) {
  int lane = threadIdx.x & 31;
  int wv   = threadIdx.x >> 5;
  int tile = blockIdx.x * WPB + wv;           // always < TILES
  int lo = lane & 15, hi = lane >> 4;
  int m = tile * 16 + lo;                     // A-matrix row (node)
  float rdeg = 1.0f / fmaxf(deg[m], 1.0f);
  const float* xrow = x   + (size_t)m * 64;
  const float* arow = agg + (size_t)m * 64;

  v8f c[4];
#pragma unroll
  for (int j = 0; j < 4; ++j) c[j] = (v8f)(0.0f);

#pragma unroll
  for (int kk = 0; kk < 16; ++kk) {
    int k = kk * 4 + 2 * hi;
    v2f a1, a2;
    a1.x = arow[k] * rdeg; a1.y = arow[k + 1] * rdeg;   // normalized aggregate
    a2.x = xrow[k];        a2.y = xrow[k + 1];          // root path
#pragma unroll
    for (int j = 0; j < 4; ++j) {
      int n = j * 16 + lo;
      v2f bw, br;
      bw.x = W[k * 64 + n]; bw.y = W[(k + 1) * 64 + n];
      br.x = R[k * 64 + n]; br.y = R[(k + 1) * 64 + n];
      c[j] = wmma4(a1, bw, c[j]);
      c[j] = wmma4(a2, br, c[j]);
    }
  }

#pragma unroll
  for (int j = 0; j < 4; ++j) {
    int n = j * 16 + lo;
    float bias = b[n];
#pragma unroll
    for (int r = 0; r < 8; ++r) {
      int mm = tile * 16 + r + 8 * hi;
      h[(size_t)mm * 64 + n] = fmaxf(c[j][r] + bias, 0.0f);
    }
  }
}

// --------------------------- mu / logvar layer -----------------------------
// mu = (agg2/deg)@Wmu + h@Rmu + bmu ; lv likewise.  Fout = 32.
__global__ void mulv_kernel(const float* __restrict__ h,
                            const float* __restrict__ agg,
                            const float* __restrict__ deg,
                            const float* __restrict__ Wmu,
                            const float* __restrict__ Rmu,
                            const float* __restrict__ bmu,
                            const float* __restrict__ Wlv,
                            const float* __restrict__ Rlv,
                            const float* __restrict__ blv,
                            float* __restrict__ mu_out,
                            float* __restrict__ lv_out) {
  int lane = threadIdx.x & 31;
  int wv   = threadIdx.x >> 5;
  int tile = blockIdx.x * WPB + wv;
  int lo = lane & 15, hi = lane >> 4;
  int m = tile * 16 + lo;
  float rdeg = 1.0f / fmaxf(deg[m], 1.0f);
  const float* hrow = h   + (size_t)m * 64;
  const float* arow = agg + (size_t)m * 64;

  v8f cmu[2], clv[2];
#pragma unroll
  for (int j = 0; j < 2; ++j) { cmu[j] = (v8f)(0.0f); clv[j] = (v8f)(0.0f); }

#pragma unroll
  for (int kk = 0; kk < 16; ++kk) {
    int k = kk * 4 + 2 * hi;
    v2f aa, ah;
    aa.x = arow[k] * rdeg; aa.y = arow[k + 1] * rdeg;
    ah.x = hrow[k];        ah.y = hrow[k + 1];
#pragma unroll
    for (int j = 0; j < 2; ++j) {
      int n = j * 16 + lo;
      v2f bwm, brm, bwl, brl;
      bwm.x = Wmu[k * 32 + n]; bwm.y = Wmu[(k + 1) * 32 + n];
      brm.x = Rmu[k * 32 + n]; brm.y = Rmu[(k + 1) * 32 + n];
      bwl.x = Wlv[k * 32 + n]; bwl.y = Wlv[(k + 1) * 32 + n];
      brl.x = Rlv[k * 32 + n]; brl.y = Rlv[(k + 1) * 32 + n];
      cmu[j] = wmma4(aa, bwm, cmu[j]);
      cmu[j] = wmma4(ah, brm, cmu[j]);
      clv[j] = wmma4(aa, bwl, clv[j]);
      clv[j] = wmma4(ah, brl, clv[j]);
    }
  }

#pragma unroll
  for (int j = 0; j < 2; ++j) {
    int n = j * 16 + lo;
    float bm = bmu[n], bl = blv[n];
#pragma unroll
    for (int r = 0; r < 8; ++r) {
      int mm = tile * 16 + r + 8 * hi;
      mu_out[(size_t)mm * 32 + n] = cmu[j][r] + bm;
      lv_out[(size_t)mm * 32 + n] = clv[j][r] + bl;
    }
  }
}

// --------------------------- decoder ---------------------------------------
// z = mu + eps*exp(0.5*lv); hd = relu(z@fc1_w + fc1_b); recon = hd@fc2_w + fc2_b
__global__ void decoder_kernel(const float* __restrict__ mu,
                               const float* __restrict__ lv,
                               const float* __restrict__ eps,
                               const float* __restrict__ fc1w,  // 32x64
                               const float* __restrict__ fc1b,
                               const float* __restrict__ fc2w,  // 64x64
                               const float* __restrict__ fc2b,
                               float* __restrict__ recon) {
  __shared__ float hd[WPB][16][66];           // 66-float pitch: no bank conflicts
  int lane = threadIdx.x & 31;
  int wv   = threadIdx.x >> 5;
  int tile = blockIdx.x * WPB + wv;
  int lo = lane & 15, hi = lane >> 4;
  int m = tile * 16 + lo;
  const float* mrow = mu  + (size_t)m * 32;
  const float* lrow = lv  + (size_t)m * 32;
  const float* erow = eps + (size_t)m * 32;

  // GEMM1: z(16x32) @ fc1w(32x64)
  v8f c[4];
#pragma unroll
  for (int j = 0; j < 4; ++j) c[j] = (v8f)(0.0f);
#pragma unroll
  for (int kk = 0; kk < 8; ++kk) {
    int k = kk * 4 + 2 * hi;
    v2f a;
    a.x = mrow[k]     + erow[k]     * expf(0.5f * lrow[k]);
    a.y = mrow[k + 1] + erow[k + 1] * expf(0.5f * lrow[k + 1]);
#pragma unroll
    for (int j = 0; j < 4; ++j) {
      int n = j * 16 + lo;
      v2f b;
      b.x = fc1w[k * 64 + n]; b.y = fc1w[(k + 1) * 64 + n];
      c[j] = wmma4(a, b, c[j]);
    }
  }
#pragma unroll
  for (int j = 0; j < 4; ++j) {
    int n = j * 16 + lo;
    float bias = fc1b[n];
#pragma unroll
    for (int r = 0; r < 8; ++r)
      hd[wv][r + 8 * hi][n] = fmaxf(c[j][r] + bias, 0.0f);
  }
  __syncthreads();

  // GEMM2: hd(16x64) @ fc2w(64x64)
  v8f c2[4];
#pragma unroll
  for (int j = 0; j < 4; ++j) c2[j] = (v8f)(0.0f);
#pragma unroll
  for (int kk = 0; kk < 16; ++kk) {
    int k = kk * 4 + 2 * hi;
    v2f a;
    a.x = hd[wv][lo][k]; a.y = hd[wv][lo][k + 1];
#pragma unroll
    for (int j = 0; j < 4; ++j) {
      int n = j * 16 + lo;
      v2f b;
      b.x = fc2w[k * 64 + n]; b.y = fc2w[(k + 1) * 64 + n];
      c2[j] = wmma4(a, b, c2[j]);
    }
  }
#pragma unroll
  for (int j = 0; j < 4; ++j) {
    int n = j * 16 + lo;
    float bias = fc2b[n];
#pragma unroll
    for (int r = 0; r < 8; ++r) {
      int mm = tile * 16 + r + 8 * hi;
      recon[(size_t)mm * 64 + n] = c2[j][r] + bias;
    }
  }
}

// --------------------------- launcher --------------------------------------

extern "C" void kernel_launch(void* const* d_in, const int* in_sizes, int n_in,
                              void* d_out, int out_size, void* d_ws, size_t ws_size,
                              hipStream_t stream) {
  const float* x       = (const float*)d_in[0];
  const int*   ei      = (const int*)d_in[1];
  const float* eps     = (const float*)d_in[2];
  const float* w1      = (const float*)d_in[3];   // [K,64,64] -> use W[0]
  const float* root1   = (const float*)d_in[4];
  const float* b1      = (const float*)d_in[5];
  const float* w_mu    = (const float*)d_in[6];   // [K,64,32] -> use W[0]
  const float* root_mu = (const float*)d_in[7];
  const float* b_mu    = (const float*)d_in[8];
  const float* w_lv    = (const float*)d_in[9];
  const float* root_lv = (const float*)d_in[10];
  const float* b_lv    = (const float*)d_in[11];
  const float* fc1w    = (const float*)d_in[12];
  const float* fc1b    = (const float*)d_in[13];
  const float* fc2w    = (const float*)d_in[14];
  const float* fc2b    = (const float*)d_in[15];

  const int* src = ei;
  const int* dst = ei + N_EDGES;

  float* out   = (float*)d_out;
  float* recon = out;                                   // [N,64]
  float* mu    = out + (size_t)N_NODES * 64;            // [N,32]
  float* lv    = mu + (size_t)N_NODES * 32;             // [N,32]

  float* ws   = (float*)d_ws;
  float* deg  = ws;                                     // [N]
  float* agg1 = deg + N_NODES;                          // [N,64]
  float* agg2 = agg1 + (size_t)N_NODES * 64;            // [N,64]
  float* h    = agg2 + (size_t)N_NODES * 64;            // [N,64]

  // 1. zero deg + agg1 + agg2 (contiguous: N*129 floats)
  {
    int n4 = (N_NODES * 129) / 4;
    int blocks = (n4 + 255) / 256;
    zero_kernel<<<blocks, 256, 0, stream>>>((float4*)ws, n4);
  }
  // 2. degree counts
  deg_kernel<<<(N_EDGES + 255) / 256, 256, 0, stream>>>(dst, deg);
  // 3. scatter x -> agg1
  scatter_kernel<<<(N_EDGES * 16) / 256, 256, 0, stream>>>(x, src, dst, agg1);
  // 4. conv layer 1 -> h
  layer1_kernel<<<GEMM_BLOCKS, GEMM_THREADS, 0, stream>>>(x, agg1, deg, w1, root1, b1, h);
  // 5. scatter h -> agg2
  scatter_kernel<<<(N_EDGES * 16) / 256, 256, 0, stream>>>(h, src, dst, agg2);
  // 6. mu / logvar
  mulv_kernel<<<GEMM_BLOCKS, GEMM_THREADS, 0, stream>>>(h, agg2, deg,
      w_mu, root_mu, b_mu, w_lv, root_lv, b_lv, mu, lv);
  // 7. decoder -> recon
  decoder_kernel<<<GEMM_BLOCKS, GEMM_THREADS, 0, stream>>>(mu, lv, eps,
      fc1w, fc1b, fc2w, fc2b, recon);
}